// composed_loss_function_32366873542744
// MI455X (gfx1250) — compile-verified
//
#include <hip/hip_runtime.h>

// sum((pred - Y_obs)^2) over N*D = 33.55M f32 elements.
// Memory-bound: 268 MB read @ 23.3 TB/s ~= 11.5 us floor.
// Pass 1: 2048 blocks stream with NT b128 loads, FMA-accumulate in f32,
//         reduce wave -> block via V_WMMA_F32_16X16X4_F32 (A x ones = row sums,
//         exact in f32) + one shfl_xor + tiny LDS fold; one partial per block.
// Pass 2: single block deterministically folds the 2048 partials -> d_out[0].

typedef float v2f __attribute__((ext_vector_type(2)));
typedef float v4f __attribute__((ext_vector_type(4)));
typedef float v8f __attribute__((ext_vector_type(8)));

#define TPB    256
#define BLOCKS 2048
#define WAVES  (TPB / 32)

// Wave32 reduction of two per-lane f32 accumulators to the wave total,
// using the all-f32 WMMA: D = A(16x4) * ones(4x16).
// A layout (ISA 7.12.2): lanes 0-15 = rows 0-15 with VGPR0=K0,VGPR1=K1;
// lanes 16-31 = rows 0-15 with VGPR0=K2,VGPR1=K3. Our {a0,a1} per lane maps
// exactly onto it; any bijective mapping is fine since we want the total sum.
// D row m (replicated across all 16 columns) = sum of row m of A.
// Lane L in 0..15 holds D[0..7][L] in its 8 D VGPRs; lanes 16..31 hold rows 8..15.
__device__ __forceinline__ float wave_sum_wmma(float a0, float a1) {
    v2f A;
    A.x = a0;
    A.y = a1;
    v2f B;
    B.x = 1.0f;
    B.y = 1.0f;
    v8f C = {};
    // (neg_a, A, neg_b, B, c_mod, C, reuse_a, reuse_b) -> v_wmma_f32_16x16x4_f32
    v8f D = __builtin_amdgcn_wmma_f32_16x16x4_f32(false, A, false, B,
                                                  (short)0, C, false, false);
    float s = ((D[0] + D[1]) + (D[2] + D[3])) + ((D[4] + D[5]) + (D[6] + D[7]));
    // lanes 0-15 hold sum of rows 0-7, lanes 16-31 hold sum of rows 8-15.
    s += __shfl_xor(s, 16, 32);
    return s; // every lane now holds the wave total
}

__global__ __launch_bounds__(TPB) void sqdiff_partial_kernel(
    const float* __restrict__ pred, const float* __restrict__ yobs,
    float* __restrict__ partial, int n4, int n) {
    const v4f* __restrict__ p4 = (const v4f*)pred;
    const v4f* __restrict__ y4 = (const v4f*)yobs;

    float a0 = 0.0f, a1 = 0.0f;
    int       i      = (int)blockIdx.x * TPB + (int)threadIdx.x;
    const int stride = BLOCKS * TPB;
#pragma unroll 4
    for (; i < n4; i += stride) {
        v4f p = __builtin_nontemporal_load(p4 + i); // global_load_b128, TH=NT
        v4f y = __builtin_nontemporal_load(y4 + i);
        float d0 = p.x - y.x;
        float d1 = p.y - y.y;
        float d2 = p.z - y.z;
        float d3 = p.w - y.w;
        a0 = fmaf(d0, d0, a0);
        a1 = fmaf(d1, d1, a1);
        a0 = fmaf(d2, d2, a0);
        a1 = fmaf(d3, d3, a1);
    }
    // Scalar tail (n not multiple of 4) — handled by one thread; reconverges
    // before the WMMA below so EXEC is all-ones there.
    if (blockIdx.x == 0 && threadIdx.x == 0) {
        for (int t = n4 * 4; t < n; ++t) {
            float d = pred[t] - yobs[t];
            a0 = fmaf(d, d, a0);
        }
    }

    float wsum = wave_sum_wmma(a0, a1);

    __shared__ float lds[WAVES];
    const int lane = (int)threadIdx.x & 31;
    const int wave = (int)threadIdx.x >> 5;
    if (lane == 0) lds[wave] = wsum;
    __syncthreads();
    if (threadIdx.x == 0) {
        float s = 0.0f;
#pragma unroll
        for (int w = 0; w < WAVES; ++w) s += lds[w];
        partial[blockIdx.x] = s;
    }
}

__global__ __launch_bounds__(TPB) void final_reduce_kernel(
    const float* __restrict__ partial, int n, float* __restrict__ out) {
    float a0 = 0.0f, a1 = 0.0f;
    for (int i = (int)threadIdx.x; i < n; i += TPB) a0 += partial[i];

    float wsum = wave_sum_wmma(a0, a1);

    __shared__ float lds[WAVES];
    const int lane = (int)threadIdx.x & 31;
    const int wave = (int)threadIdx.x >> 5;
    if (lane == 0) lds[wave] = wsum;
    __syncthreads();
    if (threadIdx.x == 0) {
        float s = 0.0f;
#pragma unroll
        for (int w = 0; w < WAVES; ++w) s += lds[w];
        out[0] = s;
    }
}

extern "C" void kernel_launch(void* const* d_in, const int* in_sizes, int n_in,
                              void* d_out, int out_size, void* d_ws, size_t ws_size,
                              hipStream_t stream) {
    const float* pred = (const float*)d_in[0];
    const float* yobs = (const float*)d_in[1];
    // d_in[2] (nan_array) is dead in the reference.
    float* out     = (float*)d_out;
    float* partial = (float*)d_ws; // BLOCKS * 4 bytes = 8 KB scratch

    const int n  = in_sizes[0]; // 8192*4096 = 33,554,432
    const int n4 = n >> 2;

    sqdiff_partial_kernel<<<BLOCKS, TPB, 0, stream>>>(pred, yobs, partial, n4, n);
    final_reduce_kernel<<<1, TPB, 0, stream>>>(partial, BLOCKS, out);
}